// LocalDualDirectedMessagePassingLayer_28913719837269
// MI455X (gfx1250) — compile-verified
//
#include <hip/hip_runtime.h>

typedef float v2f __attribute__((ext_vector_type(2)));
typedef float v8f __attribute__((ext_vector_type(8)));

namespace {
constexpr int kNNodes = 200000;
constexpr int kNDest  = 8192;
constexpr int kNEdge  = 262144;
constexpr int kDMem = 128, kDFeat = 128, kDEdge = 64, kDTime = 32;
constexpr int kDN  = 128;   // all layer output widths
constexpr int kK1  = 256;   // mem + feat
constexpr int kK2  = 224;   // read + edge + time
constexpr int kWS  = 136;   // LDS weight row stride (pad: +16-bank shift between lane halves)
constexpr int kAS1 = 260;   // LDS activation stride, 256-wide tiles
constexpr int kAS2 = 228;   // LDS activation stride, 224-wide tiles
constexpr int kChunk = 16 * kAS1 + 16 * kAS2;  // one activation buffer (floats)
constexpr int kEdgeTiles = kNEdge / 16;
constexpr int kDestTiles = kNDest / 16;
constexpr int kEdgeGrid  = 2048;
}

// ---- CDNA5 async global->LDS copy (ASYNCcnt-tracked, bypasses VGPRs) ----
__device__ __forceinline__ void async_copy_b128(const void* gptr, unsigned int lds_addr) {
  asm volatile("global_load_async_to_lds_b128 %0, %1, off"
               :: "v"(lds_addr), "v"(gptr)
               : "memory");
}
__device__ __forceinline__ void wait_async0() {
  asm volatile("s_wait_asynccnt 0" ::: "memory");
}
// LDS flat-aperture addresses map to the LDS byte offset via addr[31:0].
__device__ __forceinline__ unsigned int lds_byte_addr(const void* p) {
  return (unsigned int)(unsigned long long)p;
}

// One 16x16 output tile of (A[16xK] @ B[Kx16]) + bias via V_WMMA_F32_16X16X4_F32.
// A: row-major, stride `as`. B: column slice pre-offset to (n0 + m), row stride `bs`.
// A-fragment (16x4 f32): lanes 0-15 hold K=k,k+1 of row M=lane; lanes 16-31 hold K=k+2,k+3.
template <typename PA, typename PB>
__device__ __forceinline__ v8f wmma_rowtile(PA A, int as, PB B, int bs, int K,
                                            float bias, int m, int half) {
  v8f acc = {bias, bias, bias, bias, bias, bias, bias, bias};
  PA ap = A + m * as + 2 * half;
  PB bp = B + 2 * half * bs;
#pragma unroll 8
  for (int k = 0; k < K; k += 4) {
    v2f a, b;
    a.x = ap[0];
    a.y = ap[1];
    b.x = bp[0];
    b.y = bp[bs];
    acc = __builtin_amdgcn_wmma_f32_16x16x4_f32(false, a, false, b, (short)0, acc,
                                                false, false);
    ap += 4;
    bp += 4 * bs;
  }
  return acc;
}

// Async-issue one edge tile's activations into the LDS buffer at byte base `lbase`:
// A = concat(mem[src],feat[src]) -> lA (stride kAS1),
// edge features -> lA2 cols [128,192), time encodings -> lA2 cols [192,224).
__device__ __forceinline__ void issue_gather(
    int e0, unsigned int lbase,
    const float* __restrict__ node_memory, const float* __restrict__ node_features,
    const float* __restrict__ edge_features, const float* __restrict__ time_encoding,
    const int* __restrict__ source_ids, const int* __restrict__ edge_ids, int tid) {
  const unsigned int lA2base = lbase + 16u * kAS1 * 4u;
  // 1024 + 256 + 128 = 1408 16-byte transfers per tile.
  for (int idx = tid; idx < 1408; idx += 256) {
    const void* g;
    unsigned int l;
    if (idx < 1024) {
      const int r = idx >> 6, q = idx & 63;
      const int src = source_ids[e0 + r];
      g = (q < 32) ? (const void*)(node_memory + (size_t)src * kDMem + q * 4)
                   : (const void*)(node_features + (size_t)src * kDFeat + (q - 32) * 4);
      l = lbase + (unsigned int)(r * kAS1 + q * 4) * 4u;
    } else if (idx < 1280) {
      const int j = idx - 1024;
      const int r = j >> 4, q = j & 15;
      const int eid = edge_ids[e0 + r];
      g = edge_features + (size_t)eid * kDEdge + q * 4;
      l = lA2base + (unsigned int)(r * kAS2 + 128 + q * 4) * 4u;
    } else {
      const int j = idx - 1280;
      const int r = j >> 3, q = j & 7;
      g = time_encoding + (size_t)(e0 + r) * kDTime + q * 4;
      l = lA2base + (unsigned int)(r * kAS2 + 192 + q * 4) * 4u;
    }
    async_copy_b128(g, l);
  }
}

__global__ __launch_bounds__(256) void edge_msg_kernel(
    const float* __restrict__ node_memory, const float* __restrict__ node_features,
    const float* __restrict__ edge_features, const float* __restrict__ time_encoding,
    const int* __restrict__ source_ids, const int* __restrict__ edge_ids,
    const int* __restrict__ dest_seg,
    const float* __restrict__ W_read, const float* __restrict__ b_read,
    const float* __restrict__ W_msg, const float* __restrict__ b_msg,
    float* __restrict__ msg_sum, float* __restrict__ cnt) {
  extern __shared__ float smem[];
  float* lW1    = smem;                  // 256 x 136
  float* lW2    = lW1 + kK1 * kWS;       // 224 x 136
  float* actbuf = lW2 + kK2 * kWS;       // 2 x kChunk, double-buffered

  const int tid  = threadIdx.x;
  const int wave = tid >> 5;
  const int lane = tid & 31;
  const int half = lane >> 4;
  const int m    = lane & 15;
  const int n0   = wave * 16;

  // Stage weights once per block (padded stride for conflict-free B fragments).
  for (int idx = tid; idx < kK1 * kDN; idx += 256)
    lW1[(idx >> 7) * kWS + (idx & 127)] = W_read[idx];
  for (int idx = tid; idx < kK2 * kDN; idx += 256)
    lW2[(idx >> 7) * kWS + (idx & 127)] = W_msg[idx];

  const float bias1 = b_read[n0 + m];
  const float bias2 = b_msg[n0 + m];
  const int G = gridDim.x;
  const unsigned int actbase = lds_byte_addr(actbuf);

  // Prologue: async-prefetch first tile into buffer 0.
  if (blockIdx.x < kEdgeTiles)
    issue_gather(blockIdx.x * 16, actbase, node_memory, node_features,
                 edge_features, time_encoding, source_ids, edge_ids, tid);

  int it = 0;
  for (int t = blockIdx.x; t < kEdgeTiles; t += G, ++it) {
    // Pure pointer arithmetic off `smem` so addrspace(3) inference survives
    // and LDS traffic stays on the DS path (no flat fallback).
    const int off = (it & 1) * kChunk;
    float* lA  = actbuf + off;
    float* lA2 = lA + 16 * kAS1;
    const int e0 = t * 16;

    wait_async0();     // my async stores into LDS have landed
    __syncthreads();   // everyone's have; prev tile's GEMM2 reads are also done

    // Overlap: prefetch the next tile into the other buffer while we compute.
    if (t + G < kEdgeTiles)
      issue_gather((t + G) * 16,
                   actbase + (unsigned int)(((it + 1) & 1) * kChunk) * 4u,
                   node_memory, node_features, edge_features, time_encoding,
                   source_ids, edge_ids, tid);

    if (tid < 16) atomicAdd(&cnt[dest_seg[e0 + tid]], 1.0f);

    // GEMM1: src_read = relu(src_in @ W_read + b_read) -> lA2 cols [0,128)
    v8f acc = wmma_rowtile(lA, kAS1, lW1 + n0 + m, kWS, kK1, bias1, m, half);
    for (int r = 0; r < 8; ++r)
      lA2[(r + 8 * half) * kAS2 + n0 + m] = fmaxf(acc[r], 0.0f);
    __syncthreads();

    // GEMM2: msgs = relu(msg_in @ W_msg + b_msg); atomic segment-sum scatter.
    acc = wmma_rowtile(lA2, kAS2, lW2 + n0 + m, kWS, kK2, bias2, m, half);
    for (int r = 0; r < 8; ++r) {
      const int row = r + 8 * half;
      const int d = dest_seg[e0 + row];
      atomicAdd(&msg_sum[(size_t)d * kDN + n0 + m], fmaxf(acc[r], 0.0f));
    }
    // Next iteration's wait_async + barrier fences buffer reuse.
  }
}

__global__ __launch_bounds__(256) void dest_update_kernel(
    const float* __restrict__ node_memory, const float* __restrict__ node_features,
    const int* __restrict__ node_ids,
    const float* __restrict__ W_read, const float* __restrict__ b_read,
    const float* __restrict__ W_agg, const float* __restrict__ b_agg,
    const float* __restrict__ W_upd, const float* __restrict__ b_upd,
    const float* __restrict__ W_write, const float* __restrict__ b_write,
    const float* __restrict__ msg_sum, const float* __restrict__ cnt,
    float* __restrict__ out) {
  extern __shared__ float smem[];
  float* bufX = smem;               // 16 x 260
  float* bufY = bufX + 16 * kAS1;   // 16 x 260

  const int tid  = threadIdx.x;
  const int wave = tid >> 5;
  const int lane = tid & 31;
  const int half = lane >> 4;
  const int m    = lane & 15;
  const int n0   = wave * 16;
  const int i0   = blockIdx.x * 16;
  if (i0 >= kNDest) return;

  // bufX = dst_in = concat(node_memory[nid], node_features[nid])
  for (int idx = tid; idx < 16 * 64; idx += 256) {
    const int r = idx >> 6, q = idx & 63;
    const int nid = node_ids[i0 + r];
    const float4* p =
        (q < 32) ? ((const float4*)(node_memory + (size_t)nid * kDMem) + q)
                 : ((const float4*)(node_features + (size_t)nid * kDFeat) + (q - 32));
    *(float4*)(bufX + r * kAS1 + q * 4) = *p;
  }
  // bufY cols [128,256) = msg_mean
  for (int idx = tid; idx < 16 * kDN; idx += 256) {
    const int r = idx >> 7, c = idx & 127;
    const float cc = cnt[i0 + r];
    bufY[r * kAS1 + 128 + c] = msg_sum[(size_t)(i0 + r) * kDN + c] / fmaxf(cc, 1.0f);
  }
  __syncthreads();

  // dst_read = relu(dst_in @ W_read + b_read) -> bufY cols [0,128)
  v8f acc = wmma_rowtile(bufX, kAS1, W_read + n0 + m, kDN, kK1, b_read[n0 + m], m, half);
  for (int r = 0; r < 8; ++r)
    bufY[(r + 8 * half) * kAS1 + n0 + m] = fmaxf(acc[r], 0.0f);
  __syncthreads();

  // agg = relu([dst_read, msg_mean] @ W_agg + b_agg) -> bufX cols [0,128)
  acc = wmma_rowtile(bufY, kAS1, W_agg + n0 + m, kDN, kK1, b_agg[n0 + m], m, half);
  // carry dst_read into bufX cols [128,256) for the updater input
  for (int idx = tid; idx < 16 * kDN; idx += 256) {
    const int r = idx >> 7, c = idx & 127;
    bufX[r * kAS1 + 128 + c] = bufY[r * kAS1 + c];
  }
  for (int r = 0; r < 8; ++r)
    bufX[(r + 8 * half) * kAS1 + n0 + m] = fmaxf(acc[r], 0.0f);
  __syncthreads();

  // upd = relu([agg, dst_read] @ W_upd + b_upd) -> bufY cols [0,128)
  acc = wmma_rowtile(bufX, kAS1, W_upd + n0 + m, kDN, kK1, b_upd[n0 + m], m, half);
  for (int r = 0; r < 8; ++r)
    bufY[(r + 8 * half) * kAS1 + n0 + m] = fmaxf(acc[r], 0.0f);
  __syncthreads();

  // write = tanh(upd @ W_write + b_write), scatter into out rows node_ids[...]
  acc = wmma_rowtile(bufY, kAS1, W_write + n0 + m, kDN, kDN, b_write[n0 + m], m, half);
  for (int r = 0; r < 8; ++r) {
    const int row = r + 8 * half;
    const int nid = node_ids[i0 + row];
    out[(size_t)nid * kDMem + n0 + m] = tanhf(acc[r]);
  }
}

__global__ void copy_memory_kernel(const float* __restrict__ src,
                                   float* __restrict__ dst, int n4) {
  const float4* s = (const float4*)src;
  float4* d = (float4*)dst;
  for (int i = blockIdx.x * blockDim.x + threadIdx.x; i < n4;
       i += gridDim.x * blockDim.x)
    d[i] = s[i];
}

__global__ void zero_ws_kernel(float* __restrict__ ws, int n) {
  for (int i = blockIdx.x * blockDim.x + threadIdx.x; i < n;
       i += gridDim.x * blockDim.x)
    ws[i] = 0.0f;
}

extern "C" void kernel_launch(void* const* d_in, const int* in_sizes, int n_in,
                              void* d_out, int out_size, void* d_ws, size_t ws_size,
                              hipStream_t stream) {
  const float* node_memory   = (const float*)d_in[0];
  const float* node_features = (const float*)d_in[1];
  const float* edge_features = (const float*)d_in[2];
  const float* time_encoding = (const float*)d_in[3];
  const int*   node_ids      = (const int*)d_in[4];
  const int*   source_ids    = (const int*)d_in[5];
  const int*   edge_ids      = (const int*)d_in[6];
  const int*   dest_seg      = (const int*)d_in[7];
  const float* W_read  = (const float*)d_in[8];
  const float* b_read  = (const float*)d_in[9];
  const float* W_msg   = (const float*)d_in[10];
  const float* b_msg   = (const float*)d_in[11];
  const float* W_agg   = (const float*)d_in[12];
  const float* b_agg   = (const float*)d_in[13];
  const float* W_upd   = (const float*)d_in[14];
  const float* b_upd   = (const float*)d_in[15];
  const float* W_write = (const float*)d_in[16];
  const float* b_write = (const float*)d_in[17];
  float* out = (float*)d_out;

  float* msg_sum = (float*)d_ws;                    // [8192 x 128]
  float* cnt     = msg_sum + (size_t)kNDest * kDN;  // [8192]

  const int nzero = kNDest * kDN + kNDest;
  zero_ws_kernel<<<(nzero + 255) / 256, 256, 0, stream>>>((float*)d_ws, nzero);

  const int n4 = kNNodes * kDMem / 4;
  copy_memory_kernel<<<4096, 256, 0, stream>>>(node_memory, out, n4);

  const size_t edge_lds =
      (size_t)(kK1 * kWS + kK2 * kWS + 2 * kChunk) * sizeof(float);
  edge_msg_kernel<<<kEdgeGrid, 256, edge_lds, stream>>>(
      node_memory, node_features, edge_features, time_encoding, source_ids,
      edge_ids, dest_seg, W_read, b_read, W_msg, b_msg, msg_sum, cnt);

  const size_t dest_lds = (size_t)(2 * 16 * kAS1) * sizeof(float);
  dest_update_kernel<<<kDestTiles, 256, dest_lds, stream>>>(
      node_memory, node_features, node_ids, W_read, b_read, W_agg, b_agg, W_upd,
      b_upd, W_write, b_write, msg_sum, cnt, out);
}